// FocalLoss_72103910965956
// MI455X (gfx1250) — compile-verified
//
#include <hip/hip_runtime.h>
#include <math.h>

// Problem constants (from reference)
constexpr int B_ = 8, A_ = 49104, C_ = 90, M_ = 20, E_ = 128, NID_ = 128;

typedef __attribute__((ext_vector_type(2))) float v2f;
typedef __attribute__((ext_vector_type(8))) float v8f;

// ---------------------------------------------------------------- k_zero
__global__ void k_zero(float* cls_acc, float* reg_acc, float* emb_acc, int* npos) {
    int t = threadIdx.x;
    if (t < B_) { cls_acc[t] = 0.f; reg_acc[t] = 0.f; emb_acc[t] = 0.f; npos[t] = 0; }
}

// ---------------------------------------------------------------- k_assign
// Per (b, anchor): IoU vs M annotations -> iou_max / first-argmax / positive.
// Positives: compact into plist, accumulate CIoU regression term, count.
__global__ void __launch_bounds__(256)
k_assign(const float* __restrict__ anchors, const float* __restrict__ ann,
         const float* __restrict__ regs,
         float* __restrict__ iou_max_o, int* __restrict__ info_o,
         int* __restrict__ plist, float* __restrict__ reg_acc, int* __restrict__ npos)
{
    int b = blockIdx.y;
    int a = blockIdx.x * blockDim.x + threadIdx.x;

    __shared__ float sann[M_ * 6];
    if (threadIdx.x < M_ * 6) sann[threadIdx.x] = ann[(size_t)b * M_ * 6 + threadIdx.x];
    __syncthreads();
    if (a >= A_) return;

    // anchor row layout: [y1, x1, y2, x2]
    const float ay1 = anchors[a * 4 + 0], ax1 = anchors[a * 4 + 1];
    const float ay2 = anchors[a * 4 + 2], ax2 = anchors[a * 4 + 3];
    const float aarea = (ay2 - ay1) * (ax2 - ax1);

    float best = -__builtin_inff();
    int arg = 0;
    for (int m = 0; m < M_; ++m) {
        const float* bb = &sann[m * 6];
        float v;
        if (bb[4] == -1.0f) {
            v = -1.0f;
        } else {
            // annotation boxes are x1,y1,x2,y2
            float iw = fmaxf(fminf(ax2, bb[2]) - fmaxf(ax1, bb[0]), 0.f);
            float ih = fmaxf(fminf(ay2, bb[3]) - fmaxf(ay1, bb[1]), 0.f);
            float ua = fmaxf(aarea + (bb[2] - bb[0]) * (bb[3] - bb[1]) - iw * ih, 1e-8f);
            v = iw * ih / ua;
        }
        if (v > best) { best = v; arg = m; }   // strict > == first-occurrence argmax
    }

    iou_max_o[(size_t)b * A_ + a] = best;
    const float* ab = &sann[arg * 6];
    int lab = (int)ab[4], idt = (int)ab[5];
    info_o[(size_t)b * A_ + a] = (lab & 0xFFFF) | (idt << 16);

    if (best >= 0.5f) {
        int slot = atomicAdd(&npos[b], 1);
        plist[(size_t)b * A_ + slot] = a;

        // decode predicted box and CIoU with assigned box
        const float* rg = &regs[((size_t)b * A_ + a) * 4];
        float aw = ax2 - ax1, ah = ay2 - ay1;
        float acx = ax1 + 0.5f * aw, acy = ay1 + 0.5f * ah;
        float w = expf(rg[3]) * aw, h = expf(rg[2]) * ah;
        float yc = rg[0] * ah + acy, xc = rg[1] * aw + acx;
        float x1a = xc - 0.5f * w, y1a = yc - 0.5f * h;
        float x2a = xc + 0.5f * w, y2a = yc + 0.5f * h;
        float x1b = ab[0], y1b = ab[1], x2b = ab[2], y2b = ab[3];

        float inter = fmaxf(fminf(x2a, x2b) - fmaxf(x1a, x1b), 0.f) *
                      fmaxf(fminf(y2a, y2b) - fmaxf(y1a, y1b), 0.f);
        float a1 = (x2a - x1a) * (y2a - y1a), a2 = (x2b - x1b) * (y2b - y1b);
        float iou = inter / (a1 + a2 - inter + 1e-16f);
        float odx = fmaxf(x2a, x2b) - fminf(x1a, x1b);
        float ody = fmaxf(y2a, y2b) - fminf(y1a, y1b);
        float od = odx * odx + ody * ody;
        float w1 = x2a - x1a, h1 = y2a - y1a, w2 = x2b - x1b, h2 = y2b - y1b;
        float cdx = (x1a + 0.5f * w1) - (x1b + 0.5f * w2);
        float cdy = (y1a + 0.5f * h1) - (y1b + 0.5f * h2);
        float u = (cdx * cdx + cdy * cdy) / od;
        float dv = (atanf(w2 / h2) - atanf(w1 / h1)) * 0.63661977236758134f; // / (pi/2)
        float vv = dv * dv;
        float alpha = vv / (1.f - iou + vv + 1e-8f);
        float ci = fminf(fmaxf(iou - u - alpha * vv, -1.f), 1.f);
        atomicAdd(&reg_acc[b], ci);
    }
}

// ---------------------------------------------------------------- k_focal
// Streams classifications (141 MB). One transcendental per element.
__global__ void __launch_bounds__(256)
k_focal(const float* __restrict__ cls, const float* __restrict__ iou_max,
        const int* __restrict__ info, float* __restrict__ cls_acc)
{
    int b = blockIdx.y;
    long idx = (long)blockIdx.x * blockDim.x + threadIdx.x;
    float ls = 0.f;
    if (idx < (long)A_ * C_) {
        int a = (int)(idx / C_);
        int c = (int)(idx - (long)a * C_);
        float p = cls[(size_t)b * A_ * C_ + idx];
        p = fminf(fmaxf(p, 1e-4f), 1.f - 1e-4f);
        float im = iou_max[(size_t)b * A_ + a];
        bool pos = im >= 0.5f;
        int lab = info[(size_t)b * A_ + a] & 0xFFFF;
        // targets: -1 (ignore) / 0 / 1
        int t = -1;
        if (im < 0.4f) t = 0;
        if (pos) t = (c == lab) ? 1 : 0;
        if (t == 1) {
            float q = 1.f - p;
            ls = 0.25f * q * q * (-logf(p));
        } else if (t == 0) {
            ls = 0.75f * p * p * (-log1pf(-p));
        }
    }
    __shared__ float red[256];
    red[threadIdx.x] = ls;
    __syncthreads();
    for (int s = 128; s > 0; s >>= 1) {
        if (threadIdx.x < s) red[threadIdx.x] += red[threadIdx.x + s];
        __syncthreads();
    }
    if (threadIdx.x == 0) atomicAdd(&cls_acc[b], red[0]);
}

// ---------------------------------------------------------------- k_emb
// Per wave: 16 compacted positive rows, logits(16x128) = emb(16x128) @ id_W^T
// via V_WMMA_F32_16X16X4_F32 (32 K-steps x 8 N-tiles), then log-softmax + CE.
__global__ void __launch_bounds__(128)
k_emb(const float* __restrict__ emb, const float* __restrict__ id_W,
      const float* __restrict__ id_b, const int* __restrict__ plist,
      const int* __restrict__ info, const int* __restrict__ npos,
      float* __restrict__ emb_acc)
{
    __shared__ float sm[4 * 16 * 128];   // per-wave 16x128 logits scratch (32 KB)
    const int b    = blockIdx.y;
    const int wave = threadIdx.x >> 5;
    const int lane = threadIdx.x & 31;
    const int np   = npos[b];
    const int tile = blockIdx.x * 4 + wave;
    if (tile * 16 >= np) return;         // wave-uniform: EXEC stays all-ones for WMMA

    const int rowSel = lane & 15;        // A-matrix: lanes 0-15 / 16-31 both carry M=0..15
    const int kb     = (lane >> 4) << 1; // K-pair select: lanes<16 -> K{0,1}, >=16 -> K{2,3}

    // Gather this lane's embedding row (pad rows -> anchor 0, masked later)
    int gr   = tile * 16 + rowSel;
    int aIdx = (gr < np) ? plist[(size_t)b * A_ + gr] : 0;
    const float* erow = emb + ((size_t)b * A_ + aIdx) * E_;

    v2f afrag[32];
#pragma unroll
    for (int k = 0; k < 32; ++k) {
        afrag[k].x = erow[4 * k + kb];
        afrag[k].y = erow[4 * k + kb + 1];
    }

    float* L = &sm[wave * 16 * 128];

    for (int t = 0; t < 8; ++t) {        // N-tiles over NID=128
        const float* wrow = id_W + (size_t)(t * 16 + rowSel) * E_; // B[k,n] = id_W[n,k]
        v8f acc = {0.f, 0.f, 0.f, 0.f, 0.f, 0.f, 0.f, 0.f};
#pragma unroll
        for (int k = 0; k < 32; ++k) {
            v2f bfrag;
            bfrag.x = wrow[4 * k + kb];
            bfrag.y = wrow[4 * k + kb + 1];
            acc = __builtin_amdgcn_wmma_f32_16x16x4_f32(
                false, afrag[k], false, bfrag, (short)0, acc, false, false);
        }
        // D layout: VGPR j -> M=j (lanes 0-15) / M=j+8 (lanes 16-31), N = lane&15
        int rbase = (lane < 16) ? 0 : 8;
        int ncol  = t * 16 + rowSel;
#pragma unroll
        for (int j = 0; j < 8; ++j) L[(rbase + j) * 128 + ncol] = acc[j];
    }

    // log-softmax + CE: lanes 0-15, one row each (same-wave LDS, in-order)
    if (lane < 16) {
        int r = tile * 16 + lane;
        if (r < np) {
            int aI  = plist[(size_t)b * A_ + r];
            int idt = (info[(size_t)b * A_ + aI] >> 16) & 0xFFFF;
            const float* row = &L[lane * 128];
            float mx = -__builtin_inff();
            for (int n = 0; n < NID_; ++n) mx = fmaxf(mx, row[n] + id_b[n]);
            float s = 0.f;
            for (int n = 0; n < NID_; ++n) s += expf(row[n] + id_b[n] - mx);
            float ce = logf(s) + mx - (row[idt] + id_b[idt]);
            atomicAdd(&emb_acc[b], ce);
        }
    }
}

// ---------------------------------------------------------------- k_final
__global__ void k_final(const float* cls_acc, const float* reg_acc,
                        const float* emb_acc, const int* npos, float* out)
{
    if (threadIdx.x == 0) {
        float cs = 0.f, rs = 0.f, es = 0.f;
        for (int b = 0; b < B_; ++b) {
            float npf = fmaxf((float)npos[b], 1.f);
            bool hp = npos[b] > 0;
            cs += cls_acc[b] / npf;
            rs += hp ? (1.f - reg_acc[b] / npf) : 0.f;
            es += hp ? (emb_acc[b] / npf / npf) : 0.f;
        }
        out[0] = cs * (1.f / B_);
        out[1] = rs * (1.f / B_);
        out[2] = es * (1.f / B_);
    }
}

// ---------------------------------------------------------------- launch
extern "C" void kernel_launch(void* const* d_in, const int* in_sizes, int n_in,
                              void* d_out, int out_size, void* d_ws, size_t ws_size,
                              hipStream_t stream)
{
    const float* cls     = (const float*)d_in[0]; // (B,A,C)
    const float* regs    = (const float*)d_in[1]; // (B,A,4)
    const float* emb     = (const float*)d_in[2]; // (B,A,E)
    const float* anchors = (const float*)d_in[3]; // (1,A,4)
    const float* ann     = (const float*)d_in[4]; // (B,M,6)
    const float* id_W    = (const float*)d_in[5]; // (NID,E)
    const float* id_b    = (const float*)d_in[6]; // (NID,)
    float* out = (float*)d_out;

    // workspace carve-out (~4.7 MB)
    float* iou_max = (float*)d_ws;                       // B*A f32
    int*   info    = (int*)(iou_max + (size_t)B_ * A_);  // B*A i32
    int*   plist   = info + (size_t)B_ * A_;             // B*A i32
    float* cls_acc = (float*)(plist + (size_t)B_ * A_);  // B
    float* reg_acc = cls_acc + B_;                       // B
    float* emb_acc = reg_acc + B_;                       // B
    int*   npos    = (int*)(emb_acc + B_);               // B

    k_zero<<<1, 32, 0, stream>>>(cls_acc, reg_acc, emb_acc, npos);

    dim3 g1((A_ + 255) / 256, B_);
    k_assign<<<g1, 256, 0, stream>>>(anchors, ann, regs, iou_max, info, plist,
                                     reg_acc, npos);

    dim3 g2((int)(((long)A_ * C_ + 255) / 256), B_);
    k_focal<<<g2, 256, 0, stream>>>(cls, iou_max, info, cls_acc);

    dim3 g3(((A_ + 15) / 16 + 3) / 4, B_);   // 4 waves/block, 1 tile/wave
    k_emb<<<g3, 128, 0, stream>>>(emb, id_W, id_b, plist, info, npos, emb_acc);

    k_final<<<1, 32, 0, stream>>>(cls_acc, reg_acc, emb_acc, npos, out);
}